// BasicGroupCNN12_51797305589793
// MI455X (gfx1250) — compile-verified
//
#include <hip/hip_runtime.h>
#include <math.h>

typedef __attribute__((ext_vector_type(2))) float v2f;
typedef __attribute__((ext_vector_type(8))) float v8f;

#define HW    16384     // 128*128
#define CPAD  17424     // 132*132
#define PI_D  3.14159265358979323846

__device__ __forceinline__ int reflect_idx(int i) {
  // np.pad mode='reflect' (no edge repeat), valid for |pad| <= 3 on length-128 axis
  if (i < 0)   i = -i;
  if (i > 127) i = 254 - i;
  return i;
}

// ---------------------------------------------------------------------------
// Rotation matrices M[n][k*k][k*k]: bilinear rotation of a kxk kernel.
// ---------------------------------------------------------------------------
__global__ void rotmat_kernel(float* __restrict__ M, int k, int n) {
  int kk  = k * k;
  int idx = blockIdx.x * blockDim.x + threadIdx.x;
  if (idx >= n * kk) return;
  int m = idx / kk, ij = idx % kk;
  int i = ij / k,  j  = ij % k;
  float* row = M + (size_t)(m * kk + ij) * kk;
  for (int t = 0; t < kk; ++t) row[t] = 0.f;
  double c  = (k - 1) * 0.5;
  double th = 2.0 * PI_D * (double)m / (double)n;
  double co = cos(th), si = sin(th);
  double di = (double)i - c, dj = (double)j - c;
  double sy =  co * di + si * dj + c;
  double sx = -si * di + co * dj + c;
  int i0 = (int)floor(sy), j0 = (int)floor(sx);
  double fy = sy - i0, fx = sx - j0;
  int    iiA[4] = { i0, i0, i0 + 1, i0 + 1 };
  int    jjA[4] = { j0, j0 + 1, j0, j0 + 1 };
  double wA[4]  = { (1.0 - fy) * (1.0 - fx), (1.0 - fy) * fx, fy * (1.0 - fx), fy * fx };
  for (int t = 0; t < 4; ++t)
    if (iiA[t] >= 0 && iiA[t] < k && jjA[t] >= 0 && jjA[t] < k && wA[t] > 1e-8)
      row[iiA[t] * k + jjA[t]] = (float)wA[t];
}

// ---------------------------------------------------------------------------
// Weight transforms
// ---------------------------------------------------------------------------
// Lift weights: Wl[(m*10+cout)*3+cin][p] = sum_q lift_w[cout][cin][q] * M7[m][p][q]
__global__ void make_liftw_kernel(const float* __restrict__ lw,
                                  const float* __restrict__ M7,
                                  float* __restrict__ Wl) {
  int i = blockIdx.x * blockDim.x + threadIdx.x;
  if (i >= 80 * 3 * 49) return;
  int p   = i % 49;
  int rc  = i / 49;
  int cin = rc % 3;
  int r   = rc / 3;            // m*10 + cout
  int m = r / 10, cout = r % 10;
  const float* w  = lw + (size_t)(cout * 3 + cin) * 49;
  const float* Mr = M7 + (size_t)(m * 49 + p) * 49;
  float s = 0.f;
  for (int q = 0; q < 49; ++q) s += w[q] * Mr[q];
  Wl[i] = s;
}

// Conv weights, K padded 25->28:
// Wpad[l][r=m*10+cout][c=cin*8+o][kk] ; kk<25: rotated rolled weight, else 0
__global__ void make_convw_kernel(const float* __restrict__ cw,
                                  const float* __restrict__ M5,
                                  float* __restrict__ Wpad) {
  int i = blockIdx.x * blockDim.x + threadIdx.x;
  if (i >= 9 * 80 * 2240) return;
  int kk  = i % 28;
  int t   = i / 28;
  int c   = t % 80;
  t /= 80;
  int r   = t % 80;
  int l   = t / 80;
  if (kk >= 25) { Wpad[i] = 0.f; return; }
  int m = r / 10, cout = r % 10;
  int cin = c / 8, o = c % 8;
  int osrc = (o - m) & 7;                       // jnp.roll(W, m, axis=2)
  const float* w  = cw + ((((size_t)l * 10 + cout) * 10 + cin) * 8 + osrc) * 25;
  const float* Mr = M5 + (size_t)(m * 25 + kk) * 25;
  float s = 0.f;
  for (int q = 0; q < 25; ++q) s += w[q] * Mr[q];
  Wpad[i] = s;
}

// 1x1 group-conv weights: Wf10[r=m*16+cf][c=cin*8+o] = w10[cf][cin][(o-m)&7]
__global__ void make_w10_kernel(const float* __restrict__ w10,
                                float* __restrict__ Wf10) {
  int i = blockIdx.x * blockDim.x + threadIdx.x;
  if (i >= 128 * 80) return;
  int c = i % 80, r = i / 80;
  int m = r / 16, cf = r % 16;
  int cin = c / 8, o = c % 8;
  Wf10[i] = w10[((size_t)cf * 10 + cin) * 8 + ((o - m) & 7)];
}

// ---------------------------------------------------------------------------
// Reflect-pad the raw input (pad 3): x[2][3][128][128] -> xpad[2][3][134][134]
// ---------------------------------------------------------------------------
__global__ void padx_kernel(const float* __restrict__ x, float* __restrict__ xpad) {
  int i = blockIdx.x * blockDim.x + threadIdx.x;
  const int P = 134 * 134;
  if (i >= 6 * P) return;
  int ch = i / P, p = i % P;
  int py = p / 134, px = p % 134;
  int sy = reflect_idx(py - 3), sx = reflect_idx(px - 3);
  xpad[i] = x[((size_t)ch * 128 + sy) * 128 + sx];
}

// ---------------------------------------------------------------------------
// Lift conv (direct, 0.8 GFLOP — not worth WMMA): out remapped to act layout
// preact channel = cout*8 + m for weight row r = m*10 + cout
// ---------------------------------------------------------------------------
__global__ void lift_conv_kernel(const float* __restrict__ xpad,
                                 const float* __restrict__ Wl,
                                 float* __restrict__ preact) {
  int i = blockIdx.x * blockDim.x + threadIdx.x;
  if (i >= 2 * 80 * HW) return;
  int p = i % HW;
  int t = i / HW;
  int r = t % 80, b = t / 80;
  int y = p >> 7, x = p & 127;
  float s = 0.f;
  for (int cin = 0; cin < 3; ++cin) {
    const float* xp = xpad + ((size_t)(b * 3 + cin) * 134 + y) * 134 + x;
    const float* w  = Wl + (size_t)(r * 3 + cin) * 49;
    for (int ky = 0; ky < 7; ++ky)
      for (int kx = 0; kx < 7; ++kx)
        s += xp[ky * 134 + kx] * w[ky * 7 + kx];
  }
  int cch = (r % 10) * 8 + (r / 10);
  preact[(((size_t)b * 80 + cch) * 128 + y) * 128 + x] = s;
}

// ---------------------------------------------------------------------------
// BatchNorm statistics: group gi = 8 consecutive channels (one cout, all O).
// Emits fused scale/shift: y = x*scale + shift, then ReLU applied by consumer.
// ---------------------------------------------------------------------------
__global__ void stats_kernel(const float* __restrict__ pre,
                             const float* __restrict__ g,
                             const float* __restrict__ bb,
                             float* __restrict__ ss, int G) {
  int gi  = blockIdx.x;
  int tid = threadIdx.x;
  float s = 0.f, s2 = 0.f;
  for (int b = 0; b < 2; ++b) {
    const float* p = pre + (size_t)(b * G + gi) * 8 * HW;
    for (int i = tid; i < 8 * HW; i += 256) {
      float v = p[i];
      s += v; s2 += v * v;
    }
  }
  __shared__ float r1[256], r2[256];
  r1[tid] = s; r2[tid] = s2;
  __syncthreads();
  for (int off = 128; off > 0; off >>= 1) {
    if (tid < off) { r1[tid] += r1[tid + off]; r2[tid] += r2[tid + off]; }
    __syncthreads();
  }
  if (tid == 0) {
    const float cnt = 2.f * 8.f * (float)HW;
    float mu  = r1[0] / cnt;
    float var = r2[0] / cnt - mu * mu;
    float sc  = g[gi] * rsqrtf(var + 1e-5f);
    ss[2 * gi]     = sc;
    ss[2 * gi + 1] = bb[gi] - mu * sc;
  }
}

// ---------------------------------------------------------------------------
// BN + ReLU + reflect-pad(2): preact[2][80][128][128] -> actPad[2][80][132][132]
// ---------------------------------------------------------------------------
__global__ void bnpad_kernel(const float* __restrict__ pre,
                             const float* __restrict__ ss,
                             float* __restrict__ actPad) {
  int i = blockIdx.x * blockDim.x + threadIdx.x;
  if (i >= 2 * 80 * CPAD) return;
  int p = i % CPAD;
  int t = i / CPAD;
  int c = t % 80, b = t / 80;
  int py = p / 132, px = p % 132;
  int sy = reflect_idx(py - 2), sx = reflect_idx(px - 2);
  float v  = pre[(((size_t)b * 80 + c) * 128 + sy) * 128 + sx];
  int gi   = c >> 3;
  float y  = v * ss[2 * gi] + ss[2 * gi + 1];
  actPad[i] = y > 0.f ? y : 0.f;
}

// ---------------------------------------------------------------------------
// Core: 80->80ch 5x5 group conv as implicit GEMM with exact-fp32 WMMA.
// M=80, N=32768 pixels, K=80*28 (25 real + 3 zero-pad per input channel).
// Block: 320 thr = 10 waves = 5 M-tiles x 2 N-tiles -> 80x32 output tile.
// LDS stages the 80x5x36 padded input window (57.6 KB) once per block.
// ---------------------------------------------------------------------------
__global__ __launch_bounds__(320)
void conv80_wmma_kernel(const float* __restrict__ actPad,   // [2][80][132][132]
                        const float* __restrict__ Wpad,     // [80][80*28]
                        float* __restrict__ preact) {       // [2][80][128][128]
  extern __shared__ float smem[];                           // 80*5*36 floats
  int tile = blockIdx.x;                                    // 0..1023
  int b    = tile >> 9;
  int rem  = tile & 511;
  int y    = rem >> 2;
  int x0   = (rem & 3) << 5;

  const float* src = actPad + (size_t)b * 80 * CPAD;
  for (int i = threadIdx.x; i < 80 * 180; i += 320) {
    int c  = i / 180;
    int rr = i - c * 180;
    int ry = rr / 36;
    int rx = rr - ry * 36;
    smem[i] = src[(size_t)c * CPAD + (y + ry) * 132 + (x0 + rx)];
  }
  __syncthreads();

  int wave = threadIdx.x >> 5;
  int lane = threadIdx.x & 31;
  int mi = wave % 5;             // M tile: weight rows 16*mi ..
  int ni = wave / 5;             // N tile: pixels x0+16*ni ..
  int rr = lane & 15;
  int kh = lane >> 4;            // A/B fragment half: K offsets {0,1} or {2,3}
  int n  = ni * 16 + rr;         // B-fragment pixel column within 32-tile

  const float* Arow = Wpad + (size_t)(mi * 16 + rr) * 2240;

  v8f acc0 = {0.f, 0.f, 0.f, 0.f, 0.f, 0.f, 0.f, 0.f};
  v8f acc1 = acc0;               // two chains hide WMMA RAW latency

  for (int c = 0; c < 80; c += 2) {
    const float* A0 = Arow + c * 28;
    const float* A1 = A0 + 28;
    const float* S0 = smem + c * 180;
    const float* S1 = S0 + 180;
#pragma unroll
    for (int t = 0; t < 7; ++t) {
      int k0 = t * 4 + kh * 2;
      int k1 = k0 + 1;
      int ky0 = k0 / 5, kx0 = k0 - ky0 * 5;
      int ky1 = k1 / 5, kx1 = k1 - ky1 * 5;
      bool v0 = (k0 < 25), v1 = (k1 < 25);
      int o0 = v0 ? (ky0 * 36 + n + kx0) : 0;
      int o1 = v1 ? (ky1 * 36 + n + kx1) : 0;
      v2f a0, b0, a1, b1;
      a0.x = A0[k0];            a0.y = A0[k1];            // zero-padded tail in Wpad
      b0.x = v0 ? S0[o0] : 0.f; b0.y = v1 ? S0[o1] : 0.f;
      a1.x = A1[k0];            a1.y = A1[k1];
      b1.x = v0 ? S1[o0] : 0.f; b1.y = v1 ? S1[o1] : 0.f;
      acc0 = __builtin_amdgcn_wmma_f32_16x16x4_f32(false, a0, false, b0,
                                                   (short)0, acc0, false, false);
      acc1 = __builtin_amdgcn_wmma_f32_16x16x4_f32(false, a1, false, b1,
                                                   (short)0, acc1, false, false);
    }
  }

  // C/D layout: VGPR v: lanes 0-15 -> (M=v, N=lane); lanes 16-31 -> (M=v+8, N=lane-16)
  int N  = lane & 15;
  int Mb = (lane < 16) ? 0 : 8;
  int px = x0 + ni * 16 + N;
#pragma unroll
  for (int v = 0; v < 8; ++v) {
    float val = acc0[v] + acc1[v];
    int rflat = mi * 16 + Mb + v;                 // weight row m*10+cout
    int cch   = (rflat % 10) * 8 + (rflat / 10);  // activation channel cout*8+m
    preact[(((size_t)b * 80 + cch) * 128 + y) * 128 + px] = val;
  }
}

// ---------------------------------------------------------------------------
// Layer 10: 1x1 group conv, 80 -> 128 flat channels (reads actPad interior).
// ---------------------------------------------------------------------------
__global__ void conv1x1_kernel(const float* __restrict__ actPad,
                               const float* __restrict__ Wf10,
                               float* __restrict__ pre10) {
  int i = blockIdx.x * blockDim.x + threadIdx.x;
  if (i >= 2 * 128 * HW) return;
  int p = i % HW;
  int t = i / HW;
  int r = t % 128, b = t / 128;
  int y = p >> 7, x = p & 127;
  const float* ap = actPad + (size_t)b * 80 * CPAD + (size_t)(y + 2) * 132 + (x + 2);
  const float* w  = Wf10 + (size_t)r * 80;
  float s = 0.f;
  for (int c = 0; c < 80; ++c) s += ap[(size_t)c * CPAD] * w[c];
  int cch = (r % 16) * 8 + (r / 16);              // cf*8 + m
  pre10[(((size_t)b * 128 + cch) * 128 + y) * 128 + x] = s;
}

// ---------------------------------------------------------------------------
// Head: BN+ReLU, max over orientations, final 1x1 conv, sigmoid.
// ---------------------------------------------------------------------------
__global__ void head_kernel(const float* __restrict__ pre10,
                            const float* __restrict__ ss10,
                            const float* __restrict__ wfin,
                            float* __restrict__ out) {
  int i = blockIdx.x * blockDim.x + threadIdx.x;
  if (i >= 2 * HW) return;
  int b = i / HW, p = i % HW;
  int y = p >> 7, x = p & 127;
  float acc = 0.f;
  for (int cf = 0; cf < 16; ++cf) {
    float sc = ss10[2 * cf], sh = ss10[2 * cf + 1];
    float mx = 0.f;                               // ReLU output >= 0
    for (int m = 0; m < 8; ++m) {
      float v = pre10[(((size_t)b * 128 + cf * 8 + m) * 128 + y) * 128 + x] * sc + sh;
      v  = v > 0.f ? v : 0.f;
      mx = v > mx ? v : mx;
    }
    acc += mx * wfin[cf];
  }
  out[i] = 1.f / (1.f + __expf(-acc));
}

// ---------------------------------------------------------------------------
// Driver
// ---------------------------------------------------------------------------
extern "C" void kernel_launch(void* const* d_in, const int* in_sizes, int n_in,
                              void* d_out, int out_size, void* d_ws, size_t ws_size,
                              hipStream_t stream) {
  const float* x      = (const float*)d_in[0];
  const float* lift_w = (const float*)d_in[1];
  const float* lift_g = (const float*)d_in[2];
  const float* lift_b = (const float*)d_in[3];
  const float* conv_w = (const float*)d_in[4];
  const float* conv_g = (const float*)d_in[5];
  const float* conv_b = (const float*)d_in[6];
  const float* w10    = (const float*)d_in[7];
  const float* g10    = (const float*)d_in[8];
  const float* b10    = (const float*)d_in[9];
  const float* wfin   = (const float*)d_in[10];
  float* out = (float*)d_out;

  float* ws   = (float*)d_ws;
  float* M7   = ws;  ws += 8 * 49 * 49;
  float* M5   = ws;  ws += 8 * 25 * 25;
  float* Wl   = ws;  ws += 80 * 3 * 49;
  float* Wpad = ws;  ws += 9 * 80 * 2240;
  float* Wf10 = ws;  ws += 128 * 80;
  float* xpad = ws;  ws += 2 * 3 * 134 * 134;
  float* preA = ws;  ws += 2 * 80 * HW;
  float* actA = ws;  ws += 2 * 80 * CPAD;
  float* actB = ws;  ws += 2 * 80 * CPAD;
  float* pre10= ws;  ws += 2 * 128 * HW;
  float* ss   = ws;  ws += 64;
  // total ~56.6 MB of d_ws

  const int TB = 256;
  rotmat_kernel<<<(8 * 49 + TB - 1) / TB, TB, 0, stream>>>(M7, 7, 8);
  rotmat_kernel<<<(8 * 25 + TB - 1) / TB, TB, 0, stream>>>(M5, 5, 8);
  make_liftw_kernel<<<(80 * 3 * 49 + TB - 1) / TB, TB, 0, stream>>>(lift_w, M7, Wl);
  make_convw_kernel<<<(9 * 80 * 2240 + TB - 1) / TB, TB, 0, stream>>>(conv_w, M5, Wpad);
  make_w10_kernel<<<(128 * 80 + TB - 1) / TB, TB, 0, stream>>>(w10, Wf10);

  padx_kernel<<<(6 * 134 * 134 + TB - 1) / TB, TB, 0, stream>>>(x, xpad);
  lift_conv_kernel<<<(2 * 80 * HW + TB - 1) / TB, TB, 0, stream>>>(xpad, Wl, preA);
  stats_kernel<<<10, 256, 0, stream>>>(preA, lift_g, lift_b, ss, 10);
  bnpad_kernel<<<(2 * 80 * CPAD + TB - 1) / TB, TB, 0, stream>>>(preA, ss, actA);

  float* cur = actA;
  float* nxt = actB;
  for (int l = 0; l < 9; ++l) {
    conv80_wmma_kernel<<<1024, 320, 80 * 180 * sizeof(float), stream>>>(
        cur, Wpad + (size_t)l * 80 * 2240, preA);
    stats_kernel<<<10, 256, 0, stream>>>(preA, conv_g + l * 10, conv_b + l * 10, ss, 10);
    bnpad_kernel<<<(2 * 80 * CPAD + TB - 1) / TB, TB, 0, stream>>>(preA, ss, nxt);
    float* t = cur; cur = nxt; nxt = t;
  }

  conv1x1_kernel<<<(2 * 128 * HW + TB - 1) / TB, TB, 0, stream>>>(cur, Wf10, pre10);
  stats_kernel<<<16, 256, 0, stream>>>(pre10, g10, b10, ss, 16);
  head_kernel<<<(2 * HW + TB - 1) / TB, TB, 0, stream>>>(pre10, ss, wfin, out);
}